// GraphAttentionLayer_60000693125721
// MI455X (gfx1250) — compile-verified
//
#include <hip/hip_runtime.h>

#define N_NODES   30000
#define N_CTX     32
#define IN_FEAT   256
#define HIDDEN    128
#define ALPHA_LK  0.2f
#define NEG_INF_V -9e15f

typedef __attribute__((ext_vector_type(2))) float v2f;
typedef __attribute__((ext_vector_type(8))) float v8f;

// ---------------------------------------------------------------------------
// Kernel 1: u = W_i @ a_w[:HIDDEN]   (256-element matvec, one block)
// ---------------------------------------------------------------------------
__global__ void gat_compute_u(const float* __restrict__ W_i,
                              const float* __restrict__ a_w,
                              float* __restrict__ u) {
    const int f = threadIdx.x;            // 0..255
    const float* row = W_i + (size_t)f * HIDDEN;
    float acc = 0.f;
    #pragma unroll 8
    for (int h = 0; h < HIDDEN; ++h)
        acc += row[h] * a_w[h];
    u[f] = acc;
}

// ---------------------------------------------------------------------------
// Kernel 2: P = h_i @ W_j  via V_WMMA_F32_16X16X4_F32 (fp32-exact path).
// Grid: 1875 blocks (16 rows each) x 256 threads (8 waves; wave w -> 16 cols).
// EXEC is all-ones everywhere (exact tiling), as required for WMMA.
//
// f32 WMMA layouts (ISA 7.12.2):
//   A 16x4 : lane m = l&15, kpair kk = (l>>4)*2 ; VGPR0=K=kk, VGPR1=K=kk+1
//   B 4x16 : lane n = l&15, same half-wave K split (mirrors A)
//   C/D    : 8 VGPRs; elem r -> M = r + (l>=16 ? 8 : 0), N = l&15
// ---------------------------------------------------------------------------
__global__ void gat_gemm_P(const float* __restrict__ h,
                           const float* __restrict__ Wj,
                           float* __restrict__ P) {
    const int lane  = threadIdx.x & 31;
    const int wave  = threadIdx.x >> 5;          // 0..7 -> column tile
    const int mBase = blockIdx.x * 16;
    const int nBase = wave * 16;
    const int mn    = lane & 15;                  // M for A, N for B
    const int kk    = (lane >> 4) * 2;            // 0 or 2

    const float* aRow = h + (size_t)(mBase + mn) * IN_FEAT;

    v8f c = {};
    for (int k0 = 0; k0 < IN_FEAT; k0 += 4) {
        v2f a, b;
        a.x = aRow[k0 + kk];
        a.y = aRow[k0 + kk + 1];
        b.x = Wj[(size_t)(k0 + kk)     * HIDDEN + nBase + mn];
        b.y = Wj[(size_t)(k0 + kk + 1) * HIDDEN + nBase + mn];
        // 8 args: (neg_a, A, neg_b, B, c_mod, C, reuse_a, reuse_b)
        c = __builtin_amdgcn_wmma_f32_16x16x4_f32(
                false, a, false, b, (short)0, c, false, false);
    }

    const int rowOff = (lane >> 4) * 8;
    float* Pt = P + (size_t)(mBase + rowOff) * HIDDEN + nBase + mn;
    #pragma unroll
    for (int r = 0; r < 8; ++r)
        Pt[(size_t)r * HIDDEN] = c[r];
}

// ---------------------------------------------------------------------------
// Kernel 3: per-node scalars.  One wave32 per node.
//   s[n] = dot(h_i[n], u)          (256 floats, 8/lane)
//   q[n] = dot(P[n],  a_w[H:2H])   (128 floats, 4/lane)
// ---------------------------------------------------------------------------
__global__ void gat_row_dots(const float* __restrict__ h,
                             const float* __restrict__ P,
                             const float* __restrict__ u,
                             const float* __restrict__ a_w,
                             float* __restrict__ s,
                             float* __restrict__ q,
                             int nNodes) {
    const int lane = threadIdx.x & 31;
    const int node = blockIdx.x * (blockDim.x >> 5) + (threadIdx.x >> 5);
    if (node >= nNodes) return;

    const float4* hr = (const float4*)(h + (size_t)node * IN_FEAT);
    const float4* ur = (const float4*)u;
    float4 x0 = hr[lane],      y0 = ur[lane];
    float4 x1 = hr[lane + 32], y1 = ur[lane + 32];
    float sa = x0.x * y0.x + x0.y * y0.y + x0.z * y0.z + x0.w * y0.w
             + x1.x * y1.x + x1.y * y1.y + x1.z * y1.z + x1.w * y1.w;

    const float4* pr = (const float4*)(P + (size_t)node * HIDDEN);
    const float4* ar = (const float4*)(a_w + HIDDEN);
    float4 p = pr[lane], aa = ar[lane];
    float qa = p.x * aa.x + p.y * aa.y + p.z * aa.z + p.w * aa.w;

    #pragma unroll
    for (int off = 16; off > 0; off >>= 1) {
        sa += __shfl_xor(sa, off, 32);
        qa += __shfl_xor(qa, off, 32);
    }
    if (lane == 0) { s[node] = sa; q[node] = qa; }
}

// ---------------------------------------------------------------------------
// Kernel 4: attention + aggregation.  One wave32 per node; lane == context c.
//   score[c] = leakyrelu(s[n] + q[idx[c]] + b), masked; wave softmax;
//   out[n]   = sum_c w[c] * P[idx[c]]   (32 broadcast float4 row gathers,
//              L2-resident since P is 15.4 MB << 192 MB L2)
// ---------------------------------------------------------------------------
__global__ void gat_attention(const int* __restrict__ idx,
                              const float* __restrict__ P,
                              const float* __restrict__ s,
                              const float* __restrict__ q,
                              const float* __restrict__ a_b,
                              float* __restrict__ out,
                              int nNodes) {
    const int lane = threadIdx.x & 31;
    const int node = blockIdx.x * (blockDim.x >> 5) + (threadIdx.x >> 5);
    if (node >= nNodes) return;

    const int j = idx[(size_t)node * N_CTX + lane];

    float score;
    if (j >= 0) {
        float v = s[node] + q[j] + a_b[0];
        score = (v >= 0.f) ? v : ALPHA_LK * v;
    } else {
        score = NEG_INF_V;
    }

    // wave32 softmax over the 32 context slots
    float mx = score;
    #pragma unroll
    for (int off = 16; off > 0; off >>= 1)
        mx = fmaxf(mx, __shfl_xor(mx, off, 32));
    float e = __expf(score - mx);
    float sum = e;
    #pragma unroll
    for (int off = 16; off > 0; off >>= 1)
        sum += __shfl_xor(sum, off, 32);
    const float w = e / sum;

    // weighted row-gather accumulate; each lane owns 4 output columns
    float4 acc = make_float4(0.f, 0.f, 0.f, 0.f);
    for (int c = 0; c < N_CTX; ++c) {
        const int   jc = __shfl(j, c, 32);   // wave-uniform -> scalar branch
        const float wc = __shfl(w, c, 32);
        if (jc >= 0) {
            float4 v = ((const float4*)(P + (size_t)jc * HIDDEN))[lane];
            acc.x += wc * v.x; acc.y += wc * v.y;
            acc.z += wc * v.z; acc.w += wc * v.w;
        }
    }
    ((float4*)(out + (size_t)node * HIDDEN))[lane] = acc;
}

// ---------------------------------------------------------------------------
extern "C" void kernel_launch(void* const* d_in, const int* in_sizes, int n_in,
                              void* d_out, int out_size, void* d_ws, size_t ws_size,
                              hipStream_t stream) {
    const float* h_i = (const float*)d_in[0];   // [N, 256]
    const int*   ctx = (const int*)  d_in[1];   // [N, 32]
    const float* W_i = (const float*)d_in[2];   // [256, 128]
    const float* W_j = (const float*)d_in[3];   // [256, 128]
    const float* a_w = (const float*)d_in[4];   // [256]
    const float* a_b = (const float*)d_in[5];   // [1]
    float* out = (float*)d_out;                 // [N, 128]

    // workspace layout (floats): u[256] | s[N] | q[N] | P[N*128]  (~15.6 MB)
    float* u = (float*)d_ws;
    float* s = u + 256;
    float* q = s + N_NODES;
    float* P = q + N_NODES;

    gat_compute_u<<<1, 256, 0, stream>>>(W_i, a_w, u);
    gat_gemm_P<<<N_NODES / 16, 256, 0, stream>>>(h_i, W_j, P);
    gat_row_dots<<<(N_NODES + 7) / 8, 256, 0, stream>>>(h_i, P, u, a_w, s, q, N_NODES);
    gat_attention<<<(N_NODES + 7) / 8, 256, 0, stream>>>(ctx, P, s, q, a_b, out, N_NODES);
}